// decoder_73220602462260
// MI455X (gfx1250) — compile-verified
//
#include <hip/hip_runtime.h>
#include <hip/hip_bf16.h>

// ---------------------------------------------------------------------------
// MI455X / gfx1250 implementation. All matmuls use V_WMMA_F32_16X16X4_F32
// (fp32 in/out, matching the fp32 reference). wave32 fragment layouts per
// CDNA5 ISA 7.12.2:
//   A (16x4):  lane m=lane&15; VGPR0 = K = k0 + (lane<16 ? 0 : 2), VGPR1 = +1
//   B (4x16):  lane n=lane&15; VGPR0 = K = k0 + (lane<16 ? 0 : 2), VGPR1 = +1
//   C/D (16x16): VGPR r -> M = r + (lane<16 ? 0 : 8), N = lane&15
// LDS staging in the stage-inner kernel uses the Tensor Data Mover
// (tensor_load_to_lds + s_wait_tensorcnt), descriptors per ISA ch.8.
// The LDS object is "anchored" (volatile touch + laundered index) so the
// optimizer cannot fold the DMA-filled LDS reads to undef (the TDM builtin's
// writes to LDS are not visible to alias analysis).
// ---------------------------------------------------------------------------

typedef __attribute__((ext_vector_type(2))) float        v2f;
typedef __attribute__((ext_vector_type(8))) float        v8f;
typedef __attribute__((ext_vector_type(4))) unsigned int v4u;
typedef __attribute__((ext_vector_type(8))) int          v8i;
typedef __attribute__((ext_vector_type(4))) int          v4i;

#define Bsz 16
#define SST 256
#define SLq 48
#define VOC 16384
#define Dq  64
#define CHq 112

__device__ __forceinline__ v8f wmma4(v2f a, v2f b, v8f c) {
  return __builtin_amdgcn_wmma_f32_16x16x4_f32(
      /*neg_a=*/false, a, /*neg_b=*/false, b,
      /*c_mod=*/(short)0, c, /*reuse_a=*/false, /*reuse_b=*/false);
}

// TDM: DMA a 2D fp32 tile (tile_d0 x tile_d1) from global (row stride
// stride0 elements) into LDS at byte offset lds_off. D# packing per
// CDNA5 ISA (count=1, type=2, data_size=4B).
__device__ __forceinline__ void tdm_load_2d_f32(
    const void* gaddr, unsigned lds_off,
    unsigned tensor_d0, unsigned tensor_d1,
    unsigned tile_d0, unsigned tile_d1, unsigned stride0) {
  const unsigned long long ga = (unsigned long long)gaddr;
  v4u g0;
  g0[0] = 1u;                                           // count=1 (valid, user)
  g0[1] = lds_off;                                      // lds_addr (bytes)
  g0[2] = (unsigned)ga;                                 // global_addr[31:0]
  g0[3] = ((unsigned)(ga >> 32) & 0x01ffffffu)          // global_addr[56:32]
          | 0x80000000u;                                // type=2 ("image")
  v8i g1;
  g1[0] = 0x00020000;                                   // data_size=2 (4B)
  g1[1] = (int)((tensor_d0 & 0xffffu) << 16);           // tensor_dim0[15:0]
  g1[2] = (int)(((tensor_d0 >> 16) & 0xffffu) |
                ((tensor_d1 & 0xffffu) << 16));         // dim0[31:16]|dim1[15:0]
  g1[3] = (int)(((tensor_d1 >> 16) & 0xffffu) |
                ((tile_d0 & 0xffffu) << 16));           // dim1[31:16]|tile_dim0
  g1[4] = (int)(tile_d1 & 0xffffu);                     // tile_dim1; tile_dim2=0
  g1[5] = (int)stride0;                                 // tensor_dim0_stride lo
  g1[6] = 0;
  g1[7] = 0;
  const v4i gz4 = {0, 0, 0, 0};                         // groups 2/3 unused (2D)
  const v8i gz8 = {0, 0, 0, 0, 0, 0, 0, 0};
  __builtin_amdgcn_tensor_load_to_lds(g0, g1, gz4, gz4, gz8, 0);
}

// ---------------------------------------------------------------- gather ----
__global__ __launch_bounds__(64) void k_gather(
    const int* __restrict__ X, const float* __restrict__ EMB,
    float* __restrict__ PRED) {
  const int b = blockIdx.y;
  const int s = blockIdx.x;
  const int d = threadIdx.x;                 // 64
  const int tok = X[b * SST + s];
  PRED[((size_t)b * Dq + d) * SST + s] = EMB[(size_t)tok * Dq + d];
}

// --------------------------------------------------------------- conv1x1 ----
// Y[b,o,s] = sum_c W[o0..][wOff + c] * X[b,c,s]  (+bias)(+resid)(relu?)
// One wave per 16x16 (o,s) tile.
__global__ __launch_bounds__(32) void k_conv1x1(
    const float* __restrict__ X, const float* __restrict__ W,
    const float* __restrict__ bias, const float* __restrict__ resid,
    float* __restrict__ Y,
    int Cin, int Cout, int S, int ldW, int wOff, int doRelu) {
  const int lane = threadIdx.x & 31;
  const int n    = lane & 15;
  const int koff = (lane >> 4) << 1;          // 0 or 2
  const int half8 = (lane >> 4) << 3;         // 0 or 8
  const int tilesS = S >> 4;
  const int ts = blockIdx.x % tilesS;
  const int to = blockIdx.x / tilesS;
  const int b  = blockIdx.y;
  const int s0 = ts * 16;
  const int o0 = to * 16;
  const float* Xb = X + (size_t)b * Cin * S + s0 + n;
  const float* Wr = W + (size_t)(o0 + n) * ldW + wOff + koff;

  v8f acc = {};
  for (int k0 = 0; k0 < Cin; k0 += 4) {
    const v2f a = *(const v2f*)(Wr + k0);     // W[o][k],W[o][k+1] -> b64 load
    v2f bb;
    bb.x = Xb[(size_t)(k0 + koff) * S];
    bb.y = Xb[(size_t)(k0 + koff + 1) * S];
    acc = wmma4(a, bb, acc);
  }

  float* Yb = Y + (size_t)b * Cout * S;
  const float* Rb = resid ? resid + (size_t)b * Cout * S : nullptr;
#pragma unroll
  for (int r = 0; r < 8; r++) {
    const int mm = o0 + r + half8;
    float v = acc[r];
    if (bias) v += bias[mm];
    if (Rb)   v += Rb[(size_t)mm * S + s0 + n];
    if (doRelu) v = fmaxf(v, 0.f);
    Yb[(size_t)mm * S + s0 + n] = v;
  }
}

// ----------------------------------------------------------------- conv3 ----
// xconv[b,o,s] = relu( sum_{c,k} W[o,c,k] * t[b,c,s+k-2] + bias[o] )
// (left zero pad of 2, TPI == 0). K reduction over c*3+k = 336, via WMMA.
__global__ __launch_bounds__(32) void k_conv3(
    const float* __restrict__ T, const float* __restrict__ W,
    const float* __restrict__ bias, float* __restrict__ Y) {
  const int lane = threadIdx.x & 31;
  const int n    = lane & 15;
  const int koff = (lane >> 4) << 1;
  const int half8 = (lane >> 4) << 3;
  const int tilesS = SST >> 4;                // 16
  const int ts = blockIdx.x % tilesS;
  const int to = blockIdx.x / tilesS;         // 0..6
  const int b  = blockIdx.y;
  const int s0 = ts * 16;
  const int o0 = to * 16;
  const int K  = CHq * 3;                     // 336
  const float* Tb = T + (size_t)b * CHq * SST;
  const float* Wr = W + (size_t)(o0 + n) * K + koff;

  v8f acc = {};
  for (int k0 = 0; k0 < K; k0 += 4) {
    const int kk0 = k0 + koff;
    const int kk1 = kk0 + 1;
    const v2f a = *(const v2f*)(Wr + k0);
    const int c0 = kk0 / 3, t0 = kk0 - c0 * 3;
    const int c1 = kk1 / 3, t1 = kk1 - c1 * 3;
    const int si0 = s0 + n + t0 - 2;
    const int si1 = s0 + n + t1 - 2;
    v2f bb;
    bb.x = (si0 >= 0) ? Tb[(size_t)c0 * SST + si0] : 0.f;
    bb.y = (si1 >= 0) ? Tb[(size_t)c1 * SST + si1] : 0.f;
    acc = wmma4(a, bb, acc);
  }

  float* Yb = Y + (size_t)b * CHq * SST;
#pragma unroll
  for (int r = 0; r < 8; r++) {
    const int mm = o0 + r + half8;
    Yb[(size_t)mm * SST + s0 + n] = fmaxf(acc[r] + bias[mm], 0.f);
  }
}

// ----------------------------------------------------------- stage inner ----
// m[b,:,s] = max_l relu( W2 * relu( A[b,:,s] + E[b,:,l] ) + b2 )
// Block: 128 threads (4 waves), each wave one 16-row tile of the 64 outputs.
// A tile (64x16) and E (64x48) DMA'd into LDS by the Tensor Data Mover;
// h fragments rebuilt on the fly from LDS (no h tensor traffic at all).
__global__ __launch_bounds__(128) void k_stage_inner(
    const float* __restrict__ A, const float* __restrict__ E,
    const float* __restrict__ W2, const float* __restrict__ b2,
    float* __restrict__ M) {
  __shared__ float smem[Dq * 16 + Dq * SLq];  // [0..1023]=A tile, rest = E
  float* lA = smem;
  float* lE = smem + Dq * 16;
  const int tid = threadIdx.x;
  const int b   = blockIdx.y;
  const int s0  = blockIdx.x * 16;
  const float* Ab = A + (size_t)b * Dq * SST;
  const float* Eb = E + (size_t)b * Dq * SLq;

  if (tid < 32) {
    // A tile: rows=64 (dim1), row-chunk of 16 cols at s0 (dim0), stride SST.
    tdm_load_2d_f32(Ab + s0, 0u, SST, Dq, 16, Dq, SST);
    // E: one contiguous run of 64*48 floats.
    tdm_load_2d_f32(Eb, (unsigned)(Dq * 16 * sizeof(float)),
                    Dq * SLq, 1, Dq * SLq, 0, Dq * SLq);
    __builtin_amdgcn_s_wait_tensorcnt(0);     // TENSORcnt -> LDS data landed
    if (tid == 0) {                           // anchor: smem now has a store,
      volatile float* vp = smem;              // so LDS loads cannot be folded
      vp[0] = vp[0];                          // to undef (value unchanged)
    }
  }
  __syncthreads();                            // publish to all 4 waves

  // Laundered zero offset: opaque to the optimizer, ordered after the DMA.
  int zoff = 0;
  asm volatile("" : "+v"(zoff) :: "memory");

  const int lane = tid & 31;
  const int w    = tid >> 5;                   // o-tile 0..3
  const int n    = lane & 15;
  const int koff = (lane >> 4) << 1;
  const int half8 = (lane >> 4) << 3;
  const int o0 = w * 16;

  v2f wf[16];                                  // W2 fragments, resident
  const float* W2r = W2 + (size_t)(o0 + n) * Dq + koff;
#pragma unroll
  for (int kc = 0; kc < 16; kc++) wf[kc] = *(const v2f*)(W2r + kc * 4);

  float bias[8];
#pragma unroll
  for (int r = 0; r < 8; r++) bias[r] = b2[o0 + r + half8];

  v8f best = {};                               // relu output >= 0, so 0-init ok
  for (int l = 0; l < SLq; l++) {
    v8f acc = {};
#pragma unroll
    for (int kc = 0; kc < 16; kc++) {
      const int kk = kc * 4 + koff;
      v2f h;
      h.x = fmaxf(lA[zoff + kk * 16 + n] + lE[zoff + kk * SLq + l], 0.f);
      h.y = fmaxf(lA[zoff + (kk + 1) * 16 + n] + lE[zoff + (kk + 1) * SLq + l],
                  0.f);
      acc = wmma4(wf[kc], h, acc);
    }
#pragma unroll
    for (int r = 0; r < 8; r++) {
      best[r] = fmaxf(best[r], fmaxf(acc[r] + bias[r], 0.f));
    }
  }

  float* Mb = M + (size_t)b * Dq * SST;
#pragma unroll
  for (int r = 0; r < 8; r++)
    Mb[(size_t)(o0 + r + half8) * SST + s0 + n] = best[r];
}

// ------------------------------------------------------------ vocab GEMM ----
// out[b,s,v] = sum_d ccf[b,d,s] * emb[v,d] + fff_b[v]
// M = 4096 rows (b,s), N = 16384, K = 64. 8 waves/block; each wave keeps its
// A fragments resident (32 VGPRs) and sweeps 4 vocab tiles with them.
__global__ __launch_bounds__(256) void k_vocab(
    const float* __restrict__ CCF, const float* __restrict__ EMB,
    const float* __restrict__ FB, float* __restrict__ OUT) {
  const int tid  = threadIdx.x;
  const int lane = tid & 31;
  const int w    = tid >> 5;
  const int n    = lane & 15;
  const int koff = (lane >> 4) << 1;
  const int half8 = (lane >> 4) << 3;
  const int mt = blockIdx.y;                   // 0..255 : (b,s) row tile
  const int vtBase = blockIdx.x * 8 + w;       // 0..255
  const int row0 = mt * 16;
  const int b  = row0 >> 8;
  const int s0 = row0 & 255;
  const float* Cb = CCF + (size_t)b * Dq * SST + s0 + n;

  v2f af[16];                                  // A fragments, resident
#pragma unroll
  for (int kc = 0; kc < 16; kc++) {
    const int kk = kc * 4 + koff;
    af[kc].x = Cb[(size_t)kk * SST];           // A[m=n][k] = ccf[b][k][s0+m]
    af[kc].y = Cb[(size_t)(kk + 1) * SST];
  }

#pragma unroll
  for (int j = 0; j < 4; j++) {
    const int v0 = (vtBase + 256 * j) * 16;
    const float* Ev = EMB + (size_t)(v0 + n) * Dq + koff;
    v8f acc = {};
#pragma unroll
    for (int kc = 0; kc < 16; kc++) {
      const v2f bb = *(const v2f*)(Ev + kc * 4);  // emb[v0+n][kk..kk+1]
      acc = wmma4(af[kc], bb, acc);
    }
    const float fb = FB[v0 + n];
#pragma unroll
    for (int r = 0; r < 8; r++) {
      const int row = row0 + r + half8;
      OUT[(size_t)row * VOC + v0 + n] = acc[r] + fb;
    }
  }
}

// ------------------------------------------------------------------ host ----
extern "C" void kernel_launch(void* const* d_in, const int* in_sizes, int n_in,
                              void* d_out, int out_size, void* d_ws, size_t ws_size,
                              hipStream_t stream) {
  (void)in_sizes; (void)n_in; (void)out_size; (void)ws_size;
  const int*   x      = (const int*)  d_in[0];
  const float* ees    = (const float*)d_in[1];
  const float* emb    = (const float*)d_in[2];
  const float* cl0_w  = (const float*)d_in[3];
  const float* cl0_b  = (const float*)d_in[4];
  const float* stw[4] = {(const float*)d_in[5], (const float*)d_in[7],
                         (const float*)d_in[9], (const float*)d_in[11]};
  const float* stb[4] = {(const float*)d_in[6], (const float*)d_in[8],
                         (const float*)d_in[10], (const float*)d_in[12]};
  const float* cib_w  = (const float*)d_in[13];
  const float* cib_b  = (const float*)d_in[14];
  const float* cpr1_w = (const float*)d_in[15];   // (64,128): wt cols 0..63, we cols 64..127
  const float* cpr1_b = (const float*)d_in[16];
  const float* cpr2_w = (const float*)d_in[17];
  const float* cpr2_b = (const float*)d_in[18];
  const float* cma_w  = (const float*)d_in[19];
  const float* cma_b  = (const float*)d_in[20];
  const float* ccf_w  = (const float*)d_in[21];
  const float* ccf_b  = (const float*)d_in[22];
  const float* fff_b  = (const float*)d_in[23];

  float* out      = (float*)d_out;                               // (B,SST,VOC)
  float* out_pred = out + (size_t)Bsz * SST * VOC;               // (B,D,SST)
  float* out_ccf  = out_pred + (size_t)Bsz * Dq * SST;           // (B,D,SST)

  float* ws     = (float*)d_ws;
  float* t_buf  = ws;                                  // (B,CH,SST)
  float* xconv  = t_buf  + (size_t)Bsz * CHq * SST;    // (B,CH,SST)
  float* c_buf  = xconv  + (size_t)Bsz * CHq * SST;    // (B,D,SST)
  float* A_buf  = c_buf  + (size_t)Bsz * Dq * SST;     // (B,D,SST)
  float* E_buf  = A_buf  + (size_t)Bsz * Dq * SST;     // (B,D,SL)
  float* m_buf  = E_buf  + (size_t)Bsz * Dq * SLq;     // (B,D,SST)

  // pre_d (output #2, also feeds cl0)
  k_gather<<<dim3(SST, Bsz), 64, 0, stream>>>(x, emb, out_pred);

  // t = cl0(pre_d)
  k_conv1x1<<<dim3((SST/16)*(CHq/16), Bsz), 32, 0, stream>>>(
      out_pred, cl0_w, cl0_b, nullptr, t_buf, Dq, CHq, SST, Dq, 0, 0);

  // E = we * ees + cpr1_b   (stage-invariant — hoisted out of the loop)
  k_conv1x1<<<dim3((SLq/16)*(Dq/16), Bsz), 32, 0, stream>>>(
      ees, cpr1_w, cpr1_b, nullptr, E_buf, Dq, Dq, SLq, 2*Dq, Dq, 0);

  for (int st = 0; st < 4; st++) {
    // xconv = relu(conv3(pad2|t))
    k_conv3<<<dim3((SST/16)*(CHq/16), Bsz), 32, 0, stream>>>(
        t_buf, stw[st], stb[st], xconv);
    // c = cib(xconv)
    k_conv1x1<<<dim3((SST/16)*(Dq/16), Bsz), 32, 0, stream>>>(
        xconv, cib_w, cib_b, nullptr, c_buf, CHq, Dq, SST, CHq, 0, 0);
    // A = wt * c   (no bias; bias lives in E)
    k_conv1x1<<<dim3((SST/16)*(Dq/16), Bsz), 32, 0, stream>>>(
        c_buf, cpr1_w, nullptr, nullptr, A_buf, Dq, Dq, SST, 2*Dq, 0, 0);
    // m = max_l relu(W2 relu(A + E) + b2)   (fused, TDM + LDS resident)
    k_stage_inner<<<dim3(SST/16, Bsz), 128, 0, stream>>>(
        A_buf, E_buf, cpr2_w, cpr2_b, m_buf);
    // t = cma(m) + xconv
    k_conv1x1<<<dim3((SST/16)*(CHq/16), Bsz), 32, 0, stream>>>(
        m_buf, cma_w, cma_b, xconv, t_buf, Dq, CHq, SST, Dq, 0, 0);
  }

  // ccf_out (output #3)
  k_conv1x1<<<dim3((SST/16)*(Dq/16), Bsz), 32, 0, stream>>>(
      t_buf, ccf_w, ccf_b, nullptr, out_ccf, CHq, Dq, SST, CHq, 0, 0);

  // output #1: (b,s) x vocab GEMM + fff_b
  k_vocab<<<dim3(VOC/16/8/4, (Bsz*SST)/16), 256, 0, stream>>>(
      out_ccf, emb, fff_b, out);
}